// GraphSAGE_Module_90623809945639
// MI455X (gfx1250) — compile-verified
//
#include <hip/hip_runtime.h>
#include <hip/hip_bf16.h>
#include <math.h>

typedef float v2f __attribute__((ext_vector_type(2)));
typedef float v8f __attribute__((ext_vector_type(8)));

#define D 64
#define ACT_NONE 0
#define ACT_RELU 1
#define ACT_TANH 2

// ---------------------------------------------------------------------------
// Fused GEMM: out[N x 64] = act( A0 @ W0 (+ A1 @ W1) + bias )
// One wave per 64-row job (4 x 16-row tiles) so each B fragment load is
// reused by 4 WMMAs: per k-step 8 B-loads + 4 A-loads feed 16
// V_WMMA_F32_16X16X4_F32 ops -> matrix-pipe bound instead of vmem-issue bound.
// ---------------------------------------------------------------------------
__global__ __launch_bounds__(256)
void gemm64_kernel(const float* __restrict__ A0, const float* __restrict__ W0,
                   const float* __restrict__ A1, const float* __restrict__ W1,
                   const float* __restrict__ bias, float* __restrict__ out,
                   int nTiles16, int act)
{
    const int wave = threadIdx.x >> 5;
    const int lane = threadIdx.x & 31;
    const int nJobs = (nTiles16 + 3) >> 2;
    const int job = blockIdx.x * 8 + wave;       // wave-uniform
    if (job >= nJobs) return;

    const int tileBase = job * 4;
    const int mr = lane & 15;                    // A: row in tile / B: col in tile
    const int kh = (lane >> 4) << 1;             // 0 or 2 (K sub-pair per lane half)

    // Clamp trailing tiles (duplicate compute, stores guarded) -> no EXEC
    // divergence inside the WMMA region.
    int rows[4];
#pragma unroll
    for (int t = 0; t < 4; ++t) {
        int tt = tileBase + t;
        if (tt >= nTiles16) tt = nTiles16 - 1;
        rows[t] = tt * 16;
    }

    v8f acc[4][4];                               // [row tile][col tile]
#pragma unroll
    for (int t = 0; t < 4; ++t)
#pragma unroll
        for (int c = 0; c < 4; ++c) acc[t][c] = (v8f)0.0f;

    const int nSrc = (A1 != nullptr) ? 2 : 1;
    for (int s = 0; s < nSrc; ++s) {
        const float* __restrict__ A = s ? A1 : A0;
        const float* __restrict__ W = s ? W1 : W0;
        const float* __restrict__ aPtr[4];
#pragma unroll
        for (int t = 0; t < 4; ++t)
            aPtr[t] = A + (size_t)(rows[t] + mr) * D + kh;

#pragma unroll
        for (int k0 = 0; k0 < D; k0 += 4) {
            // B fragments: lane holds (K=k0+kh / +1, col=c*16+mr); shared by
            // all 4 row tiles. W stays hot in WGP$.
            v2f b[4];
#pragma unroll
            for (int c = 0; c < 4; ++c) {
                b[c].x = W[(size_t)(k0 + kh)     * D + c * 16 + mr];
                b[c].y = W[(size_t)(k0 + kh + 1) * D + c * 16 + mr];
            }
#pragma unroll
            for (int t = 0; t < 4; ++t) {
                const v2f a = *(const v2f*)(aPtr[t] + k0);   // aligned b64 load
#pragma unroll
                for (int c = 0; c < 4; ++c) {
                    acc[t][c] = __builtin_amdgcn_wmma_f32_16x16x4_f32(
                        /*neg_a=*/false, a, /*neg_b=*/false, b[c],
                        /*c_mod=*/(short)0, acc[t][c],
                        /*reuse_a=*/false, /*reuse_b=*/false);
                }
            }
        }
    }

    // Epilogue per C/D layout: VGPR j -> row j (lanes 0-15) / row j+8 (16-31)
    const int rowAdd = (lane >> 4) * 8;
#pragma unroll
    for (int t = 0; t < 4; ++t) {
        if (tileBase + t < nTiles16) {           // wave-uniform guard
#pragma unroll
            for (int c = 0; c < 4; ++c) {
                const int col = c * 16 + mr;
                const float bv = bias[col];
#pragma unroll
                for (int j = 0; j < 8; ++j) {
                    float v = acc[t][c][j] + bv;
                    if (act == ACT_RELU)      v = fmaxf(v, 0.0f);
                    else if (act == ACT_TANH) v = tanhf(v);
                    out[(size_t)(rows[t] + rowAdd + j) * D + col] = v;
                }
            }
        }
    }
}

// ---------------------------------------------------------------------------
// agg = 0 (also the fill value for zero-in-degree nodes, matching reference)
// ---------------------------------------------------------------------------
__global__ __launch_bounds__(256)
void zero_kernel(float4* __restrict__ ptr, int n4)
{
    int i = blockIdx.x * blockDim.x + threadIdx.x;
    if (i < n4) ptr[i] = make_float4(0.f, 0.f, 0.f, 0.f);
}

// ---------------------------------------------------------------------------
// segment_max scatter: 8 lanes per edge, 8 columns per lane.
// Messages are relu outputs (>= +0.0) so float-max == int-max on bit patterns.
// Monotone-max filter: plain load first, atomic only when strictly greater.
// A stale (lower) read just issues a redundant atomic -> race-safe; expected
// atomic count drops ~H(deg)/deg (~4.7x at avg in-degree 16).
// ---------------------------------------------------------------------------
__global__ __launch_bounds__(256)
void agg_max_kernel(const float* __restrict__ p, const int* __restrict__ src,
                    const int* __restrict__ dst, float* __restrict__ aggf, int E)
{
    long tid = (long)blockIdx.x * blockDim.x + threadIdx.x;
    int e = (int)(tid >> 3);
    if (e >= E) return;
    int cg = ((int)tid & 7) * 8;

    int s = src[e];
    int d = dst[e];
    const float4* __restrict__ ps = (const float4*)(p + (size_t)s * D + cg);
    float4 v0 = ps[0];
    float4 v1 = ps[1];

    float* af = aggf + (size_t)d * D + cg;
    float4 c0 = *(const float4*)(af);
    float4 c1 = *(const float4*)(af + 4);
    int* ad = (int*)af;
    if (v0.x > c0.x) atomicMax(ad + 0, __float_as_int(v0.x));
    if (v0.y > c0.y) atomicMax(ad + 1, __float_as_int(v0.y));
    if (v0.z > c0.z) atomicMax(ad + 2, __float_as_int(v0.z));
    if (v0.w > c0.w) atomicMax(ad + 3, __float_as_int(v0.w));
    if (v1.x > c1.x) atomicMax(ad + 4, __float_as_int(v1.x));
    if (v1.y > c1.y) atomicMax(ad + 5, __float_as_int(v1.y));
    if (v1.z > c1.z) atomicMax(ad + 6, __float_as_int(v1.z));
    if (v1.w > c1.w) atomicMax(ad + 7, __float_as_int(v1.w));
}

// ---------------------------------------------------------------------------
extern "C" void kernel_launch(void* const* d_in, const int* in_sizes, int n_in,
                              void* d_out, int out_size, void* d_ws, size_t ws_size,
                              hipStream_t stream)
{
    const float* x       = (const float*)d_in[0];
    const int*   esrc    = (const int*)  d_in[1];
    const int*   edst    = (const int*)  d_in[2];
    const float* W_pool1 = (const float*)d_in[3];
    const float* b_pool1 = (const float*)d_in[4];
    const float* W_self1 = (const float*)d_in[5];
    const float* W_neigh1= (const float*)d_in[6];
    const float* bias1   = (const float*)d_in[7];
    const float* W_pool2 = (const float*)d_in[8];
    const float* b_pool2 = (const float*)d_in[9];
    const float* W_self2 = (const float*)d_in[10];
    const float* W_neigh2= (const float*)d_in[11];
    const float* bias2   = (const float*)d_in[12];

    const int N = in_sizes[0] / D;          // 100000
    const int E = in_sizes[1];              // 1600000
    const int nTiles16 = (N + 15) / 16;
    const int nJobs = (nTiles16 + 3) / 4;

    const size_t matBytes = (size_t)N * D * sizeof(float);
    float* p   = (float*)d_ws;                               // pooled features
    float* agg = (float*)((char*)d_ws + matBytes);           // max-aggregated
    float* h1  = (float*)((char*)d_ws + 2 * matBytes);       // layer-1 output

    const dim3 blk(256);
    const int gemmBlocks = (nJobs + 7) / 8;
    const int n4         = N * D / 4;
    const int zeroBlocks = (n4 + 255) / 256;
    const long aggThr    = (long)E * 8;
    const int aggBlocks  = (int)((aggThr + 255) / 256);

    // ---- layer 1 ----
    gemm64_kernel<<<gemmBlocks, blk, 0, stream>>>(x, W_pool1, nullptr, nullptr,
                                                  b_pool1, p, nTiles16, ACT_RELU);
    zero_kernel<<<zeroBlocks, blk, 0, stream>>>((float4*)agg, n4);
    agg_max_kernel<<<aggBlocks, blk, 0, stream>>>(p, esrc, edst, agg, E);
    gemm64_kernel<<<gemmBlocks, blk, 0, stream>>>(x, W_self1, agg, W_neigh1,
                                                  bias1, h1, nTiles16, ACT_TANH);

    // ---- layer 2 ----
    gemm64_kernel<<<gemmBlocks, blk, 0, stream>>>(h1, W_pool2, nullptr, nullptr,
                                                  b_pool2, p, nTiles16, ACT_RELU);
    zero_kernel<<<zeroBlocks, blk, 0, stream>>>((float4*)agg, n4);
    agg_max_kernel<<<aggBlocks, blk, 0, stream>>>(p, esrc, edst, agg, E);
    gemm64_kernel<<<gemmBlocks, blk, 0, stream>>>(h1, W_self2, agg, W_neigh2,
                                                  bias2, (float*)d_out, nTiles16, ACT_NONE);
}